// AudioFinder_68015102100092
// MI455X (gfx1250) — compile-verified
//
#include <hip/hip_runtime.h>
#include <math.h>

typedef float v2f __attribute__((ext_vector_type(2)));
typedef float v8f __attribute__((ext_vector_type(8)));

// D(16x16,f32) += A(16x4,f32) x B(4x16,f32), full-wave (wave32) matrix op.
static __device__ __forceinline__ v8f wmma4(v2f a, v2f b, v8f c) {
    return __builtin_amdgcn_wmma_f32_16x16x4_f32(false, a, false, b, (short)0, c,
                                                 false, false);
}

#define NBATCH 16
#define CCH    80
#define WPB    8   // waves per 256-thread block

// ---------------------------------------------------------------- transpose
// in [N][T][80] -> out [N][80][T]
__global__ void af_transpose(const float* __restrict__ in, float* __restrict__ out,
                             int T) {
    size_t i = (size_t)blockIdx.x * blockDim.x + threadIdx.x;
    size_t total = (size_t)NBATCH * T * CCH;
    if (i >= total) return;
    int c = (int)(i % CCH);
    size_t nt = i / CCH;
    int t = (int)(nt % T);
    int n = (int)(nt / T);
    out[((size_t)n * CCH + c) * T + t] = in[i];
}

// ------------------------------------------------- wide conv + gated activation
// X [N][80][Tin], W [160][80][4], B [160] -> G [N][80][Tout]
// G[c][t] = tanh(Ya[c][t]) * sigmoid(Yg[c][t]),  Y = W (.) im2col(X) + B
__global__ void af_wide_gate(const float* __restrict__ X, const float* __restrict__ W,
                             const float* __restrict__ Bv, float* __restrict__ G,
                             int Tin, int Tout, int stride) {
    int wave = threadIdx.x >> 5;
    int lane = threadIdx.x & 31;
    int ttile = blockIdx.x * WPB + wave;
    if (ttile * 16 >= Tout) return;          // whole-wave uniform exit (EXEC stays full)
    int m  = lane & 15;                      // A row / B col / D col
    int kh = lane >> 4;                      // K half selector
    int octile = blockIdx.y;                 // 0..4 -> output channels octile*16..+15
    int n = blockIdx.z;

    const float* Xn = X + (size_t)n * CCH * Tin;
    int tcol = ttile * 16 + m;
    int tc   = tcol < Tout ? tcol : Tout - 1;   // clamp overhang (store is guarded)
    int oa   = octile * 16 + m;                 // a-half out channel row
    const float* Wa = W + (size_t)oa * 320 + 2 * kh;  // [o][c][k], k-slice 2kh,2kh+1
    const float* Wg = Wa + 80 * 320;                  // g-half: o + 80
    int xoff = stride * tc + 2 * kh;

    v8f acca = {0.f,0.f,0.f,0.f,0.f,0.f,0.f,0.f};
    v8f accg = {0.f,0.f,0.f,0.f,0.f,0.f,0.f,0.f};
    for (int c = 0; c < 80; ++c) {           // K = 320 -> 80 wmma steps per half
        v2f a1, a2, bb;
        a1.x = Wa[c * 4];  a1.y = Wa[c * 4 + 1];
        a2.x = Wg[c * 4];  a2.y = Wg[c * 4 + 1];
        const float* xp = Xn + (size_t)c * Tin + xoff;
        bb.x = xp[0];      bb.y = xp[1];
        acca = wmma4(a1, bb, acca);
        accg = wmma4(a2, bb, accg);
    }

    if (tcol < Tout) {
        size_t gbase = (size_t)n * CCH * Tout;
        for (int r = 0; r < 8; ++r) {
            int ch = octile * 16 + r + 8 * kh;    // D row -> channel
            float ya = acca[r] + Bv[ch];
            float yg = accg[r] + Bv[ch + 80];
            float val = tanhf(ya) * (1.0f / (1.0f + __expf(-yg)));
            G[gbase + (size_t)ch * Tout + tcol] = val;
        }
    }
}

// ------------------------------------------------------------- 1x1 conv (K=80)
// Y[c][t] = (W X)[c][t] + B[c]  (+ relu) (+ Res[c][resOff + resStride*t])
__global__ void af_conv1x1(const float* __restrict__ X, const float* __restrict__ W,
                           const float* __restrict__ Bv, const float* __restrict__ Res,
                           float* __restrict__ Y, int T, int Tres, int resOff,
                           int resStride, int doRelu) {
    int wave = threadIdx.x >> 5;
    int lane = threadIdx.x & 31;
    int ttile = blockIdx.x * WPB + wave;
    if (ttile * 16 >= T) return;
    int m = lane & 15, kh = lane >> 4;
    int octile = blockIdx.y, n = blockIdx.z;

    int tcol = ttile * 16 + m;
    int tc   = tcol < T ? tcol : T - 1;
    int o = octile * 16 + m;
    const float* Wr = W + (size_t)o * 80 + 2 * kh;   // [o][c]
    const float* Xn = X + (size_t)n * CCH * T;

    v8f acc = {0.f,0.f,0.f,0.f,0.f,0.f,0.f,0.f};
    for (int i = 0; i < 20; ++i) {                   // K = 80
        int q = 4 * i + 2 * kh;
        v2f a, b;
        a.x = Wr[4 * i];                a.y = Wr[4 * i + 1];
        b.x = Xn[(size_t)q * T + tc];   b.y = Xn[(size_t)(q + 1) * T + tc];
        acc = wmma4(a, b, acc);
    }

    if (tcol < T) {
        for (int r = 0; r < 8; ++r) {
            int ch = octile * 16 + r + 8 * kh;
            float v = acc[r] + Bv[ch];
            if (doRelu) v = fmaxf(v, 0.0f);
            if (Res)
                v += Res[(size_t)n * CCH * Tres + (size_t)ch * Tres +
                         (resOff + resStride * tcol)];
            Y[(size_t)n * CCH * T + (size_t)ch * T + tcol] = v;
        }
    }
}

// ------------------------------------------------------------------ |e|^2
__global__ void af_e2(const float* __restrict__ EMB, float* __restrict__ E2) {
    int c = blockIdx.x * blockDim.x + threadIdx.x;
    if (c >= 512) return;
    const float* e = EMB + (size_t)c * 80;
    float s = 0.f;
    for (int v = 0; v < 80; ++v) s += e[v] * e[v];
    E2[c] = s;
}

// ------------------------------------------- VQ argmin via WMMA x.e^T tiles
// score(code) = |e|^2 - 2 x.e   (|x|^2 constant over codes)
__global__ void af_vq_argmin(const float* __restrict__ ENC, const float* __restrict__ EMB,
                             const float* __restrict__ E2, int* __restrict__ IDX, int T) {
    int wave = threadIdx.x >> 5;
    int lane = threadIdx.x & 31;
    int ttile = blockIdx.x * WPB + wave;
    if (ttile * 16 >= T) return;
    int n = blockIdx.y;
    int m = lane & 15, kh = lane >> 4;
    int trow = ttile * 16 + m;
    if (trow >= T) trow = T - 1;
    const float* En = ENC + (size_t)n * CCH * T;

    float best[8];
    int   bidx[8];
    for (int r = 0; r < 8; ++r) { best[r] = 3.4e38f; bidx[r] = 0; }

    for (int ct = 0; ct < 32; ++ct) {            // 512 codes / 16
        int code = ct * 16 + m;                  // this lane's code column
        const float* er = EMB + (size_t)code * 80 + 2 * kh;
        v8f acc = {0.f,0.f,0.f,0.f,0.f,0.f,0.f,0.f};
        for (int i = 0; i < 20; ++i) {           // K = 80
            int q = 4 * i + 2 * kh;
            v2f a, b;
            a.x = En[(size_t)q * T + trow];  a.y = En[(size_t)(q + 1) * T + trow];
            b.x = er[4 * i];                 b.y = er[4 * i + 1];
            acc = wmma4(a, b, acc);
        }
        float e2c = E2[code];
        for (int r = 0; r < 8; ++r) {
            float sc = e2c - 2.0f * acc[r];
            if (sc < best[r]) { best[r] = sc; bidx[r] = code; }
            else if (sc == best[r] && code < bidx[r]) bidx[r] = code;
        }
    }

    // reduce over the 16 lanes (codes) sharing each t-row; masks <=8 stay in half
    for (int r = 0; r < 8; ++r) {
        float bv = best[r]; int bi = bidx[r];
        for (int mask = 8; mask >= 1; mask >>= 1) {
            float ov = __shfl_xor(bv, mask, 32);
            int   oi = __shfl_xor(bi, mask, 32);
            if (ov < bv || (ov == bv && oi < bi)) { bv = ov; bi = oi; }
        }
        if (m == 0) {
            int t = ttile * 16 + r + 8 * kh;     // D row -> time index
            if (t < T) IDX[(size_t)n * T + t] = bi;
        }
    }
}

// ---------------------------------------------- head: tanh((q+d)W^T+b), max_t
__global__ void af_predict(const float* __restrict__ ENCQ, const float* __restrict__ EMB,
                           const int* __restrict__ IDX, const float* __restrict__ WLIN,
                           const float* __restrict__ BLIN, float* __restrict__ OUT,
                           int T, int Tq, int qlim) {
    __shared__ float r0[256], r1[256];
    int n = blockIdx.x, tid = threadIdx.x;
    float m0 = -2.f, m1 = -2.f;
    float b0 = BLIN[0], b1 = BLIN[1];
    const float* Qn = ENCQ + (size_t)n * CCH * Tq;
    for (int t = tid; t < T; t += 256) {
        const float* e = EMB + (size_t)IDX[(size_t)n * T + t] * 80;
        float a0 = b0, a1 = b1;
        if (t < qlim) {
            int tq = t % Tq;
            for (int v = 0; v < 80; ++v) {
                float s = e[v] + Qn[(size_t)v * Tq + tq];
                a0 += s * WLIN[v];
                a1 += s * WLIN[80 + v];
            }
        } else {
            for (int v = 0; v < 80; ++v) {
                float s = e[v];
                a0 += s * WLIN[v];
                a1 += s * WLIN[80 + v];
            }
        }
        m0 = fmaxf(m0, tanhf(a0));
        m1 = fmaxf(m1, tanhf(a1));
    }
    r0[tid] = m0; r1[tid] = m1;
    __syncthreads();
    for (int s = 128; s; s >>= 1) {
        if (tid < s) {
            r0[tid] = fmaxf(r0[tid], r0[tid + s]);
            r1[tid] = fmaxf(r1[tid], r1[tid + s]);
        }
        __syncthreads();
    }
    if (tid == 0) { OUT[n * 2 + 0] = r0[0]; OUT[n * 2 + 1] = r1[0]; }
}

// =============================================================== host driver
extern "C" void kernel_launch(void* const* d_in, const int* in_sizes, int n_in,
                              void* d_out, int out_size, void* d_ws, size_t ws_size,
                              hipStream_t stream) {
    const float* search = (const float*)d_in[0];
    const float* query  = (const float*)d_in[1];
    const float* w_wide = (const float*)d_in[2];   // [4][160][80][4]
    const float* b_wide = (const float*)d_in[3];   // [4][160]
    const float* w_1x1  = (const float*)d_in[4];   // [4][80][80]
    const float* b_1x1  = (const float*)d_in[5];   // [4][80]
    const float* w_f0   = (const float*)d_in[6];   // [80][80]
    const float* b_f0   = (const float*)d_in[7];
    const float* w_f1   = (const float*)d_in[8];
    const float* b_f1   = (const float*)d_in[9];
    const float* emb    = (const float*)d_in[10];  // [512][80]
    const float* w_lin  = (const float*)d_in[11];  // [2][80]
    const float* b_lin  = (const float*)d_in[12];
    (void)in_sizes; (void)n_in; (void)out_size; (void)ws_size;

    char* ws = (char*)d_ws;
    // workspace layout (bytes)
    const size_t SZ_B0 = (size_t)NBATCH * CCH * 8192 * 4;   // 41,943,040
    const size_t SZ_B1 = (size_t)NBATCH * CCH * 4095 * 4;   // 20,966,400
    float* B0   = (float*)(ws);
    float* B1   = (float*)(ws + SZ_B0);
    float* Gb   = (float*)(ws + SZ_B0 + SZ_B1);
    float* ENCS = (float*)(ws + SZ_B0 + 2 * SZ_B1);
    float* ENCQ = (float*)(ws + SZ_B0 + 2 * SZ_B1 + (size_t)NBATCH * CCH * 2040 * 4);
    int*   IDXb = (int*)  (ws + SZ_B0 + 2 * SZ_B1 + (size_t)NBATCH * CCH * 2040 * 4
                              + (size_t)NBATCH * CCH * 504 * 4);
    float* E2b  = (float*)((char*)IDXb + (size_t)NBATCH * 2048 * 4);

    const int strides[4] = {2, 2, 1, 1};

    auto run_encoder = [&](const float* input, int T0, float* encOut) -> int {
        size_t total = (size_t)NBATCH * T0 * CCH;
        af_transpose<<<dim3((unsigned)((total + 255) / 256)), 256, 0, stream>>>(
            input, B0, T0);
        float* bufs[2] = {B0, B1};
        int cur = 0, Tin = T0;
        for (int i = 0; i < 4; ++i) {
            int s = strides[i];
            int Tout = (Tin - 4) / s + 1;
            float* Xi = bufs[cur];
            float* Xo = bufs[cur ^ 1];
            dim3 g((unsigned)((Tout + 16 * WPB - 1) / (16 * WPB)), 5, NBATCH);
            af_wide_gate<<<g, 256, 0, stream>>>(Xi, w_wide + (size_t)i * 51200,
                                                b_wide + i * 160, Gb, Tin, Tout, s);
            const float* res = (i == 0) ? nullptr : Xi;
            int skip = 4 - s;                 // (k - s) * d, d == 1
            int resOff = skip + s - 1;        // picks last phase of each stride group
            af_conv1x1<<<g, 256, 0, stream>>>(Gb, w_1x1 + (size_t)i * 6400,
                                              b_1x1 + i * 80, res, Xo, Tout, Tin,
                                              resOff, s, 0);
            cur ^= 1;
            Tin = Tout;
        }
        float* X4 = bufs[cur];
        float* F0 = bufs[cur ^ 1];
        dim3 gf((unsigned)((Tin + 16 * WPB - 1) / (16 * WPB)), 5, NBATCH);
        af_conv1x1<<<gf, 256, 0, stream>>>(X4, w_f0, b_f0, nullptr, F0, Tin, 0, 0, 0, 1);
        af_conv1x1<<<gf, 256, 0, stream>>>(F0, w_f1, b_f1, nullptr, encOut, Tin, 0, 0, 0, 0);
        return Tin;
    };

    af_e2<<<2, 256, 0, stream>>>(emb, E2b);

    int Tse = run_encoder(search, 8192, ENCS);   // -> 2040
    int Tqe = run_encoder(query, 2048, ENCQ);    // -> 504

    int ttiles = (Tse + 15) / 16;
    af_vq_argmin<<<dim3((unsigned)((ttiles + WPB - 1) / WPB), NBATCH), 256, 0, stream>>>(
        ENCS, emb, E2b, IDXb, Tse);

    int sf = Tse / Tqe;
    int qlim = sf * Tqe;
    af_predict<<<NBATCH, 256, 0, stream>>>(ENCQ, emb, IDXb, w_lin, b_lin,
                                           (float*)d_out, Tse, Tqe, qlim);
}